// RNNStub_38774964748556
// MI455X (gfx1250) — compile-verified
//
#include <hip/hip_runtime.h>

// Problem constants from the reference setup_inputs(): B=8, T=4096, D=1024, fp32.
#define B_  8
#define T_  4096
#define D_  1024
#define TC  128              // timesteps per chunk
#define NC  (T_ / TC)        // 32 chunks
#define LDS_PITCH 136        // floats per staged row (2*136 % 64 == 16 -> conflict-free)

typedef __attribute__((ext_vector_type(2))) float v2f;
typedef __attribute__((ext_vector_type(8))) float v8f;

// CDNA5 async global->LDS DMA (ASYNCcnt-tracked).  VDST = LDS byte offset VGPR,
// VADDR = 64-bit global address.  (CDNA5 ISA opcode 98.)
#define ASYNC_LOAD_B128(ldsoff, gaddr)                                   \
  asm volatile("global_load_async_to_lds_b128 %0, %1, off"               \
               :: "v"(ldsoff), "v"(gaddr) : "memory")

// ---------------------------------------------------------------------------
// Phase 1: per-chunk sums of x = a + b.  sums[b][c][d] = sum over chunk c.
// grid (D/256, NC, B), block 256.  Streams a,b once (forward order -> tail of
// the stream is resident in the 192MB L2 for the reversed phase 3).
// ---------------------------------------------------------------------------
__global__ void rnn_chunk_sums(const float* __restrict__ a,
                               const float* __restrict__ b,
                               float* __restrict__ sums) {
  const int ch = blockIdx.x * 256 + threadIdx.x;
  const int c  = blockIdx.y;
  const int bb = blockIdx.z;
  size_t base = ((size_t)bb * T_ + (size_t)c * TC) * D_ + ch;
  float s = 0.0f;
#pragma unroll 16
  for (int t = 0; t < TC; ++t) {
    size_t idx = base + (size_t)t * D_;
    s += a[idx] + b[idx];
  }
  sums[((size_t)bb * NC + c) * D_ + ch] = s;
}

// ---------------------------------------------------------------------------
// Phase 2: exclusive scan over chunk sums, seeded with s0 (tiny, ~2 MiB).
// ---------------------------------------------------------------------------
__global__ void rnn_chunk_offsets(const float* __restrict__ sums,
                                  const float* __restrict__ s0,
                                  float* __restrict__ offs) {
  const int ch = blockIdx.x * 256 + threadIdx.x;
  const int bb = blockIdx.z;
  float run = s0[(size_t)bb * D_ + ch];
  for (int c = 0; c < NC; ++c) {
    size_t idx = ((size_t)bb * NC + c) * D_ + ch;
    offs[idx] = run;
    run += sums[idx];
  }
}

// ---------------------------------------------------------------------------
// Phase 3: intra-chunk inclusive scan via WMMA, operands staged into LDS with
// double-buffered async global->LDS b128 DMA.
//   D(16x16) = L(16x16 lower-tri ones) x X(16 time x 16 chan) + C(carry),
//   as 4 chained V_WMMA_F32_16X16X4_F32 (exact fp32).
// Block = 256 threads (8 waves); each wave owns 16 channels of the block's
// 128-channel tile and walks its chunk in 16-step sub-tiles.
// Block indices are fully reversed vs phase 1 for L2 reuse; out is stored
// non-temporal so it does not evict a/b from L2.
// ---------------------------------------------------------------------------
__global__ void rnn_scan_wmma(const float* __restrict__ a,
                              const float* __restrict__ b,
                              const float* __restrict__ offs,
                              float* __restrict__ out) {
  __shared__ float tileA[2][16 * LDS_PITCH];
  __shared__ float tileB[2][16 * LDS_PITCH];

  const int lane = threadIdx.x & 31;
  const int wave = threadIdx.x >> 5;
  const int n    = lane & 15;          // column (channel) this lane holds in B/C/D
  const int hi   = (lane >> 4) & 1;    // 0: lanes 0-15, 1: lanes 16-31

  // Reverse the linear traversal order of phase 1 -> phase 3 starts on the
  // data phase 1 most recently left in L2.
  const int xR = gridDim.x - 1 - blockIdx.x;
  const int c  = gridDim.y - 1 - blockIdx.y;
  const int bb = gridDim.z - 1 - blockIdx.z;
  const int ch0 = xR * 128;
  const int ch  = ch0 + wave * 16 + n;

  // A = lower-triangular ones, 16x4 fp32 blocks (documented A layout):
  //   lanes 0-15: M=lane, VGPR0=K0, VGPR1=K1; lanes16-31: M=lane-16, K2/K3.
  const int m = n;
  v2f aL[4];
#pragma unroll
  for (int blk = 0; blk < 4; ++blk) {
    int k0 = 4 * blk + 2 * hi;
    aL[blk].x = (k0     <= m) ? 1.0f : 0.0f;
    aL[blk].y = (k0 + 1 <= m) ? 1.0f : 0.0f;
  }

  // Async-copy one 16(time) x 128(chan) tile of a and b into LDS buffer `buf`.
  // 4 x b128 per thread => per-wave ASYNCcnt += 4 per step.  Each wave's 32
  // lanes cover one 512B-contiguous row segment per instruction.
  auto issue_tile = [&](int buf, int tbase) {
#pragma unroll
    for (int h = 0; h < 2; ++h) {
      int id = threadIdx.x + h * 256;      // 512 chunks of 4 floats
      int r  = id >> 5;                    // row 0..15
      int c4 = (id & 31) << 2;             // col 0..124 step 4
      size_t g = ((size_t)bb * T_ + (size_t)(tbase + r)) * D_ + ch0 + c4;
      unsigned la = (unsigned)(size_t)&tileA[buf][r * LDS_PITCH + c4];
      unsigned lb = (unsigned)(size_t)&tileB[buf][r * LDS_PITCH + c4];
      ASYNC_LOAD_B128(la, (unsigned long long)(a + g));
      ASYNC_LOAD_B128(lb, (unsigned long long)(b + g));
    }
  };

  // Running state entering this chunk (s0 + previous chunks), per channel.
  float carry = offs[((size_t)bb * NC + c) * D_ + ch];

  issue_tile(0, c * TC);                   // prologue: stage step 0

  for (int s = 0; s < TC / 16; ++s) {
    const int tbase = c * TC + 16 * s;
    const int buf   = s & 1;

    if (s + 1 < TC / 16) {
      issue_tile(buf ^ 1, tbase + 16);     // prefetch next sub-tile
      asm volatile("s_wait_asynccnt 0x4" ::: "memory");  // step-s DMAs done
    } else {
      asm volatile("s_wait_asynccnt 0x0" ::: "memory");
    }
    __syncthreads();                       // all waves' DMAs for buf landed

    // C[m][n] = carry[n] for all rows (documented C/D layout).
    v8f acc;
#pragma unroll
    for (int i = 0; i < 8; ++i) acc[i] = carry;

#pragma unroll
    for (int blk = 0; blk < 4; ++blk) {
      // B-block (4x16): B[k][n] = x[tbase+4*blk+k][ch]; mirror-of-A layout:
      // lane holds column N=n, VGPR0 = row k0 (=2*hi), VGPR1 = row k0+1.
      int k0 = 4 * blk + 2 * hi;
      int l  = k0 * LDS_PITCH + wave * 16 + n;
      v2f bX;
      bX.x = tileA[buf][l]             + tileB[buf][l];
      bX.y = tileA[buf][l + LDS_PITCH] + tileB[buf][l + LDS_PITCH];
      acc = __builtin_amdgcn_wmma_f32_16x16x4_f32(
          /*neg_a=*/false, aL[blk], /*neg_b=*/false, bX,
          /*c_mod=*/(short)0, acc, /*reuse_a=*/false, /*reuse_b=*/false);
    }

    // Store the 16x16 scan tile, non-temporal (out is never re-read; keep
    // L2 for the a/b stream).
#pragma unroll
    for (int v = 0; v < 8; ++v) {
      int row = tbase + v + 8 * hi;
      __builtin_nontemporal_store(acc[v],
          &out[((size_t)bb * T_ + (size_t)row) * D_ + ch]);
    }

    // carry = row 15 of D = acc[7] on lanes 16-31, column = lane-16.
    int cbits = __builtin_amdgcn_ds_bpermute((16 + n) << 2,
                                             __float_as_int(acc[7]));
    carry = __int_as_float(cbits);

    __syncthreads();                       // done reading buf before reuse
  }
}

// ---------------------------------------------------------------------------
extern "C" void kernel_launch(void* const* d_in, const int* in_sizes, int n_in,
                              void* d_out, int out_size, void* d_ws, size_t ws_size,
                              hipStream_t stream) {
  const float* a  = (const float*)d_in[0];
  const float* b  = (const float*)d_in[1];
  const float* s0 = (const float*)d_in[2];
  float* out  = (float*)d_out;

  float* sums = (float*)d_ws;                         // B*NC*D floats (1 MiB)
  float* offs = sums + (size_t)B_ * NC * D_;          // B*NC*D floats (1 MiB)

  dim3 blk(256);
  rnn_chunk_sums   <<<dim3(D_ / 256, NC, B_), blk, 0, stream>>>(a, b, sums);
  rnn_chunk_offsets<<<dim3(D_ / 256, 1,  B_), blk, 0, stream>>>(sums, s0, offs);
  rnn_scan_wmma    <<<dim3(D_ / 128, NC, B_), blk, 0, stream>>>(a, b, offs, out);
}